// CrossScalePeriodicFeatureAggregator_57939108823220
// MI455X (gfx1250) — compile-verified
//
#include <hip/hip_runtime.h>

typedef float v2f __attribute__((ext_vector_type(2)));
typedef float v4f __attribute__((ext_vector_type(4)));
typedef float v8f __attribute__((ext_vector_type(8)));

#define AS1 __attribute__((address_space(1)))

#define SEQ    336
#define DMODEL 512
#define NBLK   4                     // 128 rows per expert / 32-row blocks
#define NWG    (4 * NBLK * SEQ)      // 5376 workgroups

// Force global (addrspace 1) vector loads — the expert-pointer select defeats
// clang's address-space inference and would otherwise emit flat_load_*.
__device__ __forceinline__ v2f gload2(const float* p) {
  return *(const AS1 v2f*)(unsigned long long)p;
}
__device__ __forceinline__ v4f gload4(const float* p) {
  return *(const AS1 v4f*)(unsigned long long)p;
}
__device__ __forceinline__ void gatomic_add(float* p, float v) {
#if __has_builtin(__builtin_amdgcn_global_atomic_fadd_f32)
  __builtin_amdgcn_global_atomic_fadd_f32((AS1 float*)(unsigned long long)p, v);
#else
  unsafeAtomicAdd(p, v);
#endif
}

// Fused MoE patch-expert + projection + gated scatter.
// One workgroup handles (expert e, 32-row n-block nb, output token t):
//   S[32,512] = X[32,512] @ W_e[j*512:(j+1)*512, :]^T + b_e[j*512:]   (j = t % p_e)
//   P[32,512] = S @ Wp^T + bp
//   out[batch_index[row], t, :] += gate[row] * P[row, :]
__global__ __launch_bounds__(256) void moe_fused_kernel(
    const float* __restrict__ xs0, const float* __restrict__ W0, const float* __restrict__ b0,
    const float* __restrict__ xs1, const float* __restrict__ W1, const float* __restrict__ b1,
    const float* __restrict__ xs2, const float* __restrict__ W2, const float* __restrict__ b2,
    const float* __restrict__ xs3, const float* __restrict__ W3, const float* __restrict__ b3,
    const float* __restrict__ gates, const float* __restrict__ Wp, const float* __restrict__ bp,
    const int* __restrict__ bidx, const int* __restrict__ eidx,
    float* __restrict__ out)
{
  __shared__ float ldsX[32 * DMODEL];   // 64 KB: input row block
  __shared__ float ldsS[32 * DMODEL];   // 64 KB: intermediate S
  __shared__ float ldsGate[32];
  __shared__ int   ldsB[32];

  const int tid  = threadIdx.x;
  const int lane = tid & 31;
  const int wave = tid >> 5;           // 0..7 (each owns 64 N columns)
  const int lr   = lane & 15;          // row/col-in-tile
  const int lh   = lane >> 4;          // lane half (selects K pair)

  int blk = blockIdx.x;
  int e   = blk / (NBLK * SEQ);
  int rem = blk % (NBLK * SEQ);
  int nb  = rem / SEQ;
  int t   = rem % SEQ;

  const int p = (e == 0) ? 4  : (e == 1) ? 8  : (e == 2) ? 12 : 24;
  const int L = (e == 0) ? 84 : (e == 1) ? 42 : (e == 2) ? 28 : 14;
  const int l = t / p;
  const int j = t % p;

  const float* xs = (e == 0) ? xs0 : (e == 1) ? xs1 : (e == 2) ? xs2 : xs3;
  const float* W  = (e == 0) ? W0  : (e == 1) ? W1  : (e == 2) ? W2  : W3;
  const float* bE = (e == 0) ? b0  : (e == 1) ? b1  : (e == 2) ? b2  : b3;

  // ---- stage X block [32][512] into LDS (each thread: 16 x float4) ----
  const float* xbase = xs + ((size_t)(nb * 32) * L + l) * DMODEL;
  #pragma unroll
  for (int it = 0; it < 16; ++it) {
    int f4 = tid + it * 256;          // float4 index within [32*512/4]
    int f  = f4 * 4;
    int m  = f >> 9;                  // / 512
    int k  = f & 511;
    v4f v = gload4(xbase + (size_t)m * L * DMODEL + k);
    *(v4f*)(&ldsX[m * DMODEL + k]) = v;
  }
  if (tid < 32) {
    int ng = e * 128 + nb * 32 + tid;             // expert-major NNZ row
    int bi = bidx[ng];
    ldsB[tid]    = bi;
    ldsGate[tid] = gates[bi * 4 + eidx[ng]];
  }
  __syncthreads();

  // ---- GEMM 1: S = X @ W_j^T + b_j  (M=32, N=512, K=512) ----
  {
    const float* Wj = W + (size_t)j * DMODEL * DMODEL;
    v8f acc[2][4];
    const float* bptr[4];
    #pragma unroll
    for (int tl = 0; tl < 4; ++tl) {
      int n0 = wave * 64 + tl * 16;
      float bv = bE[j * DMODEL + n0 + lr];
      v8f init = {bv, bv, bv, bv, bv, bv, bv, bv};
      acc[0][tl] = init;
      acc[1][tl] = init;
      bptr[tl]   = Wj + (size_t)(n0 + lr) * DMODEL + lh * 2;
    }
    const float* aptr0 = &ldsX[lr * DMODEL + lh * 2];
    const float* aptr1 = &ldsX[(16 + lr) * DMODEL + lh * 2];
    #pragma unroll 4
    for (int k0 = 0; k0 < DMODEL; k0 += 4) {
      v2f a0 = *(const v2f*)aptr0; aptr0 += 4;         // ds_load_b64 (A frag M0)
      v2f a1 = *(const v2f*)aptr1; aptr1 += 4;         // ds_load_b64 (A frag M1)
      #pragma unroll
      for (int tl = 0; tl < 4; ++tl) {
        v2f b = gload2(bptr[tl]); bptr[tl] += 4;       // global_load_b64, reused 2x
        acc[0][tl] = __builtin_amdgcn_wmma_f32_16x16x4_f32(
            false, a0, false, b, (short)0, acc[0][tl], false, false);
        acc[1][tl] = __builtin_amdgcn_wmma_f32_16x16x4_f32(
            false, a1, false, b, (short)0, acc[1][tl], false, false);
      }
    }
    // spill S tiles to LDS (C/D layout: elem r of lane -> M = r + 8*lh, N = n0+lr)
    #pragma unroll
    for (int mt = 0; mt < 2; ++mt) {
      #pragma unroll
      for (int tl = 0; tl < 4; ++tl) {
        int n0 = wave * 64 + tl * 16;
        #pragma unroll
        for (int rr = 0; rr < 8; ++rr) {
          int m = mt * 16 + rr + lh * 8;
          ldsS[m * DMODEL + n0 + lr] = acc[mt][tl][rr];
        }
      }
    }
  }
  __syncthreads();

  // ---- GEMM 2: P = S @ Wp^T + bp, gate, atomic scatter-add ----
  {
    v8f acc[2][4];
    const float* bptr[4];
    #pragma unroll
    for (int tl = 0; tl < 4; ++tl) {
      int n0 = wave * 64 + tl * 16;
      float bv = bp[n0 + lr];
      v8f init = {bv, bv, bv, bv, bv, bv, bv, bv};
      acc[0][tl] = init;
      acc[1][tl] = init;
      bptr[tl]   = Wp + (size_t)(n0 + lr) * DMODEL + lh * 2;
    }
    const float* aptr0 = &ldsS[lr * DMODEL + lh * 2];
    const float* aptr1 = &ldsS[(16 + lr) * DMODEL + lh * 2];
    #pragma unroll 4
    for (int k0 = 0; k0 < DMODEL; k0 += 4) {
      v2f a0 = *(const v2f*)aptr0; aptr0 += 4;
      v2f a1 = *(const v2f*)aptr1; aptr1 += 4;
      #pragma unroll
      for (int tl = 0; tl < 4; ++tl) {
        v2f b = gload2(bptr[tl]); bptr[tl] += 4;
        acc[0][tl] = __builtin_amdgcn_wmma_f32_16x16x4_f32(
            false, a0, false, b, (short)0, acc[0][tl], false, false);
        acc[1][tl] = __builtin_amdgcn_wmma_f32_16x16x4_f32(
            false, a1, false, b, (short)0, acc[1][tl], false, false);
      }
    }
    #pragma unroll
    for (int mt = 0; mt < 2; ++mt) {
      #pragma unroll
      for (int tl = 0; tl < 4; ++tl) {
        int n0 = wave * 64 + tl * 16;
        #pragma unroll
        for (int rr = 0; rr < 8; ++rr) {
          int m = mt * 16 + rr + lh * 8;           // row within 32-row block
          float v = acc[mt][tl][rr] * ldsGate[m];
          size_t off = (size_t)ldsB[m] * (SEQ * DMODEL) + (size_t)t * DMODEL + n0 + lr;
          gatomic_add(&out[off], v);               // global_atomic_add_f32
        }
      }
    }
  }
}

extern "C" void kernel_launch(void* const* d_in, const int* in_sizes, int n_in,
                              void* d_out, int out_size, void* d_ws, size_t ws_size,
                              hipStream_t stream) {
  (void)in_sizes; (void)n_in; (void)d_ws; (void)ws_size;
  const float* xs0 = (const float*)d_in[0];
  const float* W0  = (const float*)d_in[1];
  const float* b0  = (const float*)d_in[2];
  const float* xs1 = (const float*)d_in[3];
  const float* W1  = (const float*)d_in[4];
  const float* b1  = (const float*)d_in[5];
  const float* xs2 = (const float*)d_in[6];
  const float* W2  = (const float*)d_in[7];
  const float* b2  = (const float*)d_in[8];
  const float* xs3 = (const float*)d_in[9];
  const float* W3  = (const float*)d_in[10];
  const float* b3  = (const float*)d_in[11];
  const float* gates = (const float*)d_in[12];
  const float* Wp    = (const float*)d_in[13];
  const float* bp    = (const float*)d_in[14];
  const int*   bidx  = (const int*)d_in[15];
  const int*   eidx  = (const int*)d_in[16];
  float* out = (float*)d_out;

  // Output is accumulated with atomics: zero it every call (graph-capture safe).
  hipMemsetAsync(out, 0, (size_t)out_size * sizeof(float), stream);

  dim3 grid(NWG), block(256);
  hipLaunchKernelGGL(moe_fused_kernel, grid, block, 0, stream,
                     xs0, W0, b0, xs1, W1, b1, xs2, W2, b2, xs3, W3, b3,
                     gates, Wp, bp, bidx, eidx, out);
}